// DiMNet_36721970381413
// MI455X (gfx1250) — compile-verified
//
#include <hip/hip_runtime.h>
#include <math.h>

// ---------------------------------------------------------------------------
// DiMNet graph-attention layer for MI455X (gfx1250, wave32).
//
// Decomposition: concat(x,t) @ W  ==  x @ W_top + t @ W_bot, so all dense math
// becomes per-NODE / per-RELATION 128-deep GEMMs executed with
// V_WMMA_F32_16X16X4_F32 (exact fp32 matrix path). Edge phase is gather +
// wave-dot + f32 atomics; all hot tables (20MB node projections, 20MB
// accumulators) fit in the 192MB L2.
// ---------------------------------------------------------------------------

typedef __attribute__((ext_vector_type(2))) float v2f;
typedef __attribute__((ext_vector_type(8))) float v8f;

#define D 128
#define SELF_PAD 512  // relation table padded rows (502 rels + 1 self + pad)

// ---------------------------------------------------------------------------
// C[M x 128] = A[M x 128] @ W[128 x 128] (+ bias).  Requires M % 16 == 0.
// One wave -> one 16x16 C tile. Block = 256 threads = 8 waves = 8 col tiles.
// K loop fully unrolled: 32 x v_wmma_f32_16x16x4_f32.
//
// ISA VGPR layouts (CDNA5 ISA):
//   A 16x4 f32 : lane l (r=l%16, g=l/16), component j holds A[r][2g+j]
//   B 4x16 f32 : lane l, component j holds B[2g+j][r]
//   C 16x16 f32: lane l, component j holds C[j+8g][r]
// ---------------------------------------------------------------------------
__global__ __launch_bounds__(256) void gemm128_wmma(
    const float* __restrict__ A, const float* __restrict__ W,
    const float* __restrict__ bias, float* __restrict__ C, int M) {
  const int lane = threadIdx.x & 31;
  const int wave = threadIdx.x >> 5;
  const int r = lane & 15;   // row-in-tile (A), col-in-tile (B/C)
  const int g = lane >> 4;   // half-wave select
  const int m0 = blockIdx.x * 16;
  const int n0 = wave * 16;

  const float* __restrict__ Arow = A + (size_t)(m0 + r) * D;
  const float* __restrict__ Wcol = W + (n0 + r);

  v8f acc = {};
#pragma unroll
  for (int k = 0; k < D; k += 4) {
    const int kk = k + 2 * g;
    v2f a;
    a.x = Arow[kk];
    a.y = Arow[kk + 1];
    v2f b;
    b.x = Wcol[(size_t)kk * D];
    b.y = Wcol[(size_t)(kk + 1) * D];
    acc = __builtin_amdgcn_wmma_f32_16x16x4_f32(
        /*neg_a=*/false, a, /*neg_b=*/false, b,
        /*c_mod=*/(short)0, acc, /*reuse_a=*/false, /*reuse_b=*/false);
  }

  const float bv = bias ? bias[n0 + r] : 0.0f;
#pragma unroll
  for (int j = 0; j < 8; ++j) {
    const int row = m0 + j + 8 * g;
    C[(size_t)row * D + n0 + r] = acc[j] + bv;
  }
}

// ---------------------------------------------------------------------------
// Init: zero feat accumulators (feat_sta lives directly in d_out[0]),
// set per-node running max to -inf, sums/counts to 0.
// ---------------------------------------------------------------------------
__global__ void init_kernel(float* out0, float* featdyn, float* mdyn,
                            float* msta, float* sdyn, float* ssta,
                            float* cnts, int N) {
  const int i = blockIdx.x * 256 + threadIdx.x;
  const int ND = N * D;
  if (i < ND) {
    out0[i] = 0.0f;
    featdyn[i] = 0.0f;
  }
  if (i < N) {
    mdyn[i] = -__builtin_inff();
    msta[i] = -__builtin_inff();
    sdyn[i] = 0.0f;
    ssta[i] = 0.0f;
    cnts[i] = 0.0f;
  }
}

// Padded relation-embedding table: rows [0,R) = relemb, row R = self-relation,
// rest zero.
__global__ void build_trel(const float* __restrict__ relemb,
                           const float* __restrict__ selfrel,
                           float* __restrict__ Trel, int R) {
  const int i = blockIdx.x * 256 + threadIdx.x;
  if (i >= SELF_PAD * D) return;
  const int row = i >> 7;
  const int c = i & (D - 1);
  float v = 0.0f;
  if (row < R) v = relemb[i];
  else if (row == R) v = selfrel[c];
  Trel[i] = v;
}

// float atomic-max via signed/unsigned integer atomics (IEEE monotone trick).
__device__ inline void atomicMaxF(float* addr, float v) {
  if (v >= 0.0f)
    atomicMax((int*)addr, __float_as_int(v));
  else
    atomicMin((unsigned int*)addr, __float_as_uint(v));
}

// ---------------------------------------------------------------------------
// Pass 1: one wave per edge. attn = dot(Xq[dst]+Tq[t], Xk[src]+Tk[t]) / sqrt(D)
// plus running segment max (attn and -attn) and degree counts.
// ---------------------------------------------------------------------------
__global__ __launch_bounds__(256) void edge_attn_kernel(
    const int* __restrict__ ei, const int* __restrict__ et, int E, int N,
    int R, const float* __restrict__ Xq, const float* __restrict__ Xk,
    const float* __restrict__ Tq, const float* __restrict__ Tk,
    float* __restrict__ attn, float* __restrict__ mdyn,
    float* __restrict__ msta, float* __restrict__ cnts) {
  const int e = blockIdx.x * 8 + (threadIdx.x >> 5);
  const int lane = threadIdx.x & 31;
  const int ET = E + N;
  if (e >= ET) return;

  int s, d, t;
  if (e < E) {
    s = ei[e];
    d = ei[E + e];
    t = et[e];
  } else {
    s = d = e - E;
    t = R;  // dedicated self-relation row
  }

  const float4 q0 = ((const float4*)(Xq + (size_t)d * D))[lane];
  const float4 q1 = ((const float4*)(Tq + (size_t)t * D))[lane];
  const float4 k0 = ((const float4*)(Xk + (size_t)s * D))[lane];
  const float4 k1 = ((const float4*)(Tk + (size_t)t * D))[lane];

  float a = (q0.x + q1.x) * (k0.x + k1.x) + (q0.y + q1.y) * (k0.y + k1.y) +
            (q0.z + q1.z) * (k0.z + k1.z) + (q0.w + q1.w) * (k0.w + k1.w);
#pragma unroll
  for (int off = 16; off >= 1; off >>= 1) a += __shfl_xor(a, off, 32);

  if (lane == 0) {
    a *= 0.08838834764831845f;  // 1/sqrt(128)
    attn[e] = a;
    atomicMaxF(&mdyn[d], a);
    atomicMaxF(&msta[d], -a);
    atomicAdd(&cnts[d], 1.0f);
  }
}

// Pass 2: exp(attn - max) per edge for both softmaxes + segment sums.
__global__ void edge_expsum_kernel(const int* __restrict__ ei, int E, int N,
                                   const float* __restrict__ attn,
                                   const float* __restrict__ mdyn,
                                   const float* __restrict__ msta,
                                   float* __restrict__ edyn,
                                   float* __restrict__ esta,
                                   float* __restrict__ sdyn,
                                   float* __restrict__ ssta) {
  const int e = blockIdx.x * 256 + threadIdx.x;
  const int ET = E + N;
  if (e >= ET) return;
  const int d = (e < E) ? ei[E + e] : (e - E);
  const float a = attn[e];
  const float ed = __expf(a - mdyn[d]);
  const float es = __expf(-a - msta[d]);
  edyn[e] = ed;
  esta[e] = es;
  atomicAdd(&sdyn[d], ed);
  atomicAdd(&ssta[d], es);
}

// Pass 3: one wave per edge, scatter prob * v into the two feature
// accumulators (scatter-mean denominator folded into the edge weight).
__global__ __launch_bounds__(256) void edge_scatter_kernel(
    const int* __restrict__ ei, int E, int N, const float* __restrict__ edyn,
    const float* __restrict__ esta, const float* __restrict__ sdyn,
    const float* __restrict__ ssta, const float* __restrict__ cnts,
    const float* __restrict__ Vv, float* __restrict__ featsta,
    float* __restrict__ featdyn) {
  const int e = blockIdx.x * 8 + (threadIdx.x >> 5);
  const int lane = threadIdx.x & 31;
  const int ET = E + N;
  if (e >= ET) return;

  int s, d;
  if (e < E) {
    s = ei[e];
    d = ei[E + e];
  } else {
    s = d = e - E;
  }

  const float denom = fmaxf(cnts[d], 1.0f);
  const float wd = edyn[e] / (sdyn[d] * denom);
  const float ws = esta[e] / (ssta[d] * denom);

  const float4 v4 = ((const float4*)(Vv + (size_t)s * D))[lane];
  float* fd = featdyn + (size_t)d * D + lane * 4;
  float* fs = featsta + (size_t)d * D + lane * 4;
  atomicAdd(fs + 0, ws * v4.x);
  atomicAdd(fs + 1, ws * v4.y);
  atomicAdd(fs + 2, ws * v4.z);
  atomicAdd(fs + 3, ws * v4.w);
  atomicAdd(fd + 0, wd * v4.x);
  atomicAdd(fd + 1, wd * v4.y);
  atomicAdd(fd + 2, wd * v4.z);
  atomicAdd(fd + 3, wd * v4.w);
}

// ---------------------------------------------------------------------------
extern "C" void kernel_launch(void* const* d_in, const int* in_sizes, int n_in,
                              void* d_out, int out_size, void* d_ws,
                              size_t ws_size, hipStream_t stream) {
  const float* curr = (const float*)d_in[0];
  const float* last = (const float*)d_in[1];
  const float* relemb = (const float*)d_in[2];
  const float* selfrel = (const float*)d_in[3];
  const float* Wq = (const float*)d_in[4];
  const float* bq = (const float*)d_in[5];
  const float* Wk = (const float*)d_in[6];
  const float* bk = (const float*)d_in[7];
  const float* Wv = (const float*)d_in[8];
  const float* bv = (const float*)d_in[9];
  const float* dynW = (const float*)d_in[10];
  const int* ei = (const int*)d_in[11];
  const int* et = (const int*)d_in[12];

  const int N = in_sizes[0] / D;   // 40000 (multiple of 16)
  const int E = in_sizes[12];      // 400000
  const int R = in_sizes[2] / D;   // 502
  const int ET = E + N;            // edges incl. self-loops
  const size_t ND = (size_t)N * D;

  // Workspace carve-up (floats). ~89 MB total.
  float* ws = (float*)d_ws;
  float* Xq = ws;                     // N x D : curr @ Wq_top
  float* Xk = Xq + ND;                // N x D : last @ Wk_top
  float* Vv = Xk + ND;                // N x D : last @ Wv + bv
  float* featdyn = Vv + ND;           // N x D accumulator
  float* Trel = featdyn + ND;         // SELF_PAD x D
  float* Tq = Trel + SELF_PAD * D;    // SELF_PAD x D : Trel @ Wq_bot + bq
  float* Tk = Tq + SELF_PAD * D;      // SELF_PAD x D : Trel @ Wk_bot + bk
  float* attn = Tk + SELF_PAD * D;    // ET
  float* edyn = attn + ET;            // ET
  float* esta = edyn + ET;            // ET
  float* mdyn = esta + ET;            // N
  float* msta = mdyn + N;             // N
  float* sdyn = msta + N;             // N
  float* ssta = sdyn + N;             // N
  float* cnts = ssta + N;             // N

  float* out0 = (float*)d_out;        // feat_sta accumulates here directly
  float* out1 = out0 + ND;            // feat_dyn @ dyn_weight

  init_kernel<<<(int)((ND + 255) / 256), 256, 0, stream>>>(
      out0, featdyn, mdyn, msta, sdyn, ssta, cnts, N);
  build_trel<<<(SELF_PAD * D + 255) / 256, 256, 0, stream>>>(relemb, selfrel,
                                                             Trel, R);

  // Dense projections (WMMA fp32). Wq/Wk are (2D, D): top half multiplies the
  // node features, bottom half the relation embedding; biases fold into T*.
  gemm128_wmma<<<N / 16, 256, 0, stream>>>(curr, Wq, nullptr, Xq, N);
  gemm128_wmma<<<N / 16, 256, 0, stream>>>(last, Wk, nullptr, Xk, N);
  gemm128_wmma<<<N / 16, 256, 0, stream>>>(last, Wv, bv, Vv, N);
  gemm128_wmma<<<SELF_PAD / 16, 256, 0, stream>>>(Trel, Wq + D * D, bq, Tq,
                                                  SELF_PAD);
  gemm128_wmma<<<SELF_PAD / 16, 256, 0, stream>>>(Trel, Wk + D * D, bk, Tk,
                                                  SELF_PAD);

  // Edge phase.
  edge_attn_kernel<<<(ET + 7) / 8, 256, 0, stream>>>(
      ei, et, E, N, R, Xq, Xk, Tq, Tk, attn, mdyn, msta, cnts);
  edge_expsum_kernel<<<(ET + 255) / 256, 256, 0, stream>>>(
      ei, E, N, attn, mdyn, msta, edyn, esta, sdyn, ssta);
  edge_scatter_kernel<<<(ET + 7) / 8, 256, 0, stream>>>(
      ei, E, N, edyn, esta, sdyn, ssta, cnts, Vv, out0, featdyn);

  // out[1] = feat_dyn @ dyn_weight (WMMA fp32).
  gemm128_wmma<<<N / 16, 256, 0, stream>>>(featdyn, dynW, nullptr, out1, N);
}